// ROIAlign_42966852829221
// MI455X (gfx1250) — compile-verified
//
#include <hip/hip_runtime.h>
#include <stdint.h>

typedef uint32_t u32;
typedef __attribute__((ext_vector_type(4))) u32 u32x4;
typedef __attribute__((ext_vector_type(8))) int i32x8;
typedef __attribute__((ext_vector_type(4))) int i32x4;

#define SCALE   0.0625f
#define C_TOTAL 256
#define CCHUNK  64           // channels per pipeline stage (== blockDim)
#define NCHUNK  4
#define FH      64
#define FW      64
#define NS      14           // 7*2 sample points per axis
#define PATCH   9            // provable max footprint per axis
#define CHS     (PATCH*PATCH)   // 81 (odd -> conflict-free lane stride)
#define NBIN    49

// ---- TDM descriptor helpers (bit layout per CDNA5 ISA D# groups) ----
__device__ __forceinline__ void tdm_build(u32x4* g0, i32x8* g1, i32x4* g2, i32x4* g3,
                                          uint64_t ga, u32 lds,
                                          u32 td0, u32 td1, u32 td2,
                                          u32 t0, u32 t1, u32 t2,
                                          u32 s0, u32 s1)
{
  (*g0)[0] = 1u;                                  // count=1, user descriptor
  (*g0)[1] = lds;                                 // LDS byte address
  (*g0)[2] = (u32)ga;                             // global_addr lo
  (*g0)[3] = (u32)(ga >> 32) | 0x80000000u;       // global_addr hi | type=2
  (*g1)[0] = (int)(2u << 16);                     // wg_mask=0, data_size=4B
  (*g1)[1] = (int)((td0 & 0xFFFFu) << 16);        // tensor_dim0[15:0]
  (*g1)[2] = (int)((td0 >> 16) | ((td1 & 0xFFFFu) << 16));
  (*g1)[3] = (int)((td1 >> 16) | (t0 << 16));     // td1 hi | tile_dim0
  (*g1)[4] = (int)(t1 | (t2 << 16));              // tile_dim1 | tile_dim2
  (*g1)[5] = (int)s0;                             // dim0 stride lo32
  (*g1)[6] = (int)((s1 & 0xFFFFu) << 16);         // s0 hi16=0 | s1[15:0]
  (*g1)[7] = (int)(s1 >> 16);                     // s1[47:16]
  (*g2)[0] = (int)td2;                            // tensor_dim2
  (*g2)[1] = 0; (*g2)[2] = 0; (*g2)[3] = 0;       // no dim3, iterate off
  (*g3)[0] = 0; (*g3)[1] = 0; (*g3)[2] = 0; (*g3)[3] = 0;
}

__device__ __forceinline__ void tdm_load(u32x4 g0, i32x8 g1, i32x4 g2, i32x4 g3) {
#if defined(__clang_major__) && (__clang_major__ >= 23)
  i32x8 gx = {0, 0, 0, 0, 0, 0, 0, 0};
  __builtin_amdgcn_tensor_load_to_lds(g0, g1, g2, g3, gx, 0);
#else
  __builtin_amdgcn_tensor_load_to_lds(g0, g1, g2, g3, 0);
#endif
}

__device__ __forceinline__ void tdm_store(u32x4 g0, i32x8 g1, i32x4 g2, i32x4 g3) {
#if defined(__clang_major__) && (__clang_major__ >= 23)
  i32x8 gx = {0, 0, 0, 0, 0, 0, 0, 0};
  __builtin_amdgcn_tensor_store_from_lds(g0, g1, g2, g3, gx, 0);
#else
  __builtin_amdgcn_tensor_store_from_lds(g0, g1, g2, g3, 0);
#endif
}

__device__ __forceinline__ void interp_axis(float c, int size,
                                            int* lo, int* hi,
                                            float* w0, float* w1) {
  bool valid = (c >= -1.0f) && (c <= (float)size);
  float cc = fmaxf(c, 0.0f);
  int low = (int)floorf(cc);
  bool at_end = low >= size - 1;
  low = at_end ? size - 1 : low;
  int high = at_end ? size - 1 : low + 1;
  cc = at_end ? (float)low : cc;
  float f = cc - (float)low;
  float v = valid ? 1.0f : 0.0f;
  *lo = low; *hi = high;
  *w0 = v * (1.0f - f);   // weight for low index (validity folded in)
  *w1 = v * f;            // weight for high index
}

__global__ __launch_bounds__(CCHUNK) void roi_align_tdm_kernel(
    const float* __restrict__ feat, const float* __restrict__ rois,
    float* __restrict__ out, int K)
{
  __shared__ float buf[2][CCHUNK * CHS];          // 2 x 20736 B (double-buffered patch)
  __shared__ float obuf[CCHUNK * NBIN];           // 12544 B  (staged output chunk)
  __shared__ int   s_yl[NS], s_yh[NS], s_xl[NS], s_xh[NS];
  __shared__ float s_wy0[NS], s_wy1[NS], s_wx0[NS], s_wx1[NS];

  const int k = blockIdx.x;
  if (k >= K) return;
  const int t = (int)threadIdx.x;

  const float* r = rois + (size_t)k * 5;
  const int   bidx = (int)r[0];
  const float x1 = r[1] * SCALE - 0.5f;
  const float y1 = r[2] * SCALE - 0.5f;
  const float x2 = r[3] * SCALE - 0.5f;
  const float y2 = r[4] * SCALE - 0.5f;
  const float bh = (y2 - y1) * (1.0f / 7.0f);
  const float bw = (x2 - x1) * (1.0f / 7.0f);

  if (t < NS) {
    float c = y1 + ((float)t + 0.5f) * (bh * 0.5f);           // sr = 2
    int lo, hi; float w0, w1;
    interp_axis(c, FH, &lo, &hi, &w0, &w1);
    s_yl[t] = lo; s_yh[t] = hi; s_wy0[t] = w0; s_wy1[t] = w1;
  } else if (t < 2 * NS) {
    int i = t - NS;
    float c = x1 + ((float)i + 0.5f) * (bw * 0.5f);
    int lo, hi; float w0, w1;
    interp_axis(c, FW, &lo, &hi, &w0, &w1);
    s_xl[i] = lo; s_xh[i] = hi; s_wx0[i] = w0; s_wx1[i] = w1;
  }
  __syncthreads();

  // Patch origin: sample coords are monotone, so min index is entry 0.
  const int y0 = __builtin_amdgcn_readfirstlane(s_yl[0]);
  const int x0 = __builtin_amdgcn_readfirstlane(s_xl[0]);
  const int bb = __builtin_amdgcn_readfirstlane(bidx);

  const uint64_t feat_b = (uint64_t)(uintptr_t)feat
                        + 4ull * ((uint64_t)bb * C_TOTAL * (FH * FW)
                                  + (uint64_t)y0 * FW + (uint64_t)x0);
  const u32 td0 = (u32)(FW - x0);     // remaining x extent -> OOB zero-fill
  const u32 td1 = (u32)(FH - y0);     // remaining y extent
  const u32 lds_b0 = (u32)(uintptr_t)(void*)&buf[0][0];
  const u32 lds_b1 = (u32)(uintptr_t)(void*)&buf[1][0];
  const u32 lds_ob = (u32)(uintptr_t)(void*)&obuf[0];
  const bool w0lane = (t < 32);       // wave 0 drives the TDM engine

  // Prologue: kick off chunk 0's patch DMA.
  if (w0lane) {
    u32x4 g0; i32x8 g1; i32x4 g2, g3;
    tdm_build(&g0, &g1, &g2, &g3, feat_b, lds_b0, td0, td1, CCHUNK,
              PATCH, PATCH, CCHUNK, FW, FH * FW);
    tdm_load(g0, g1, g2, g3);
  }

  for (int q = 0; q < NCHUNK; ++q) {
    const u32 cur_lds = (q & 1) ? lds_b1 : lds_b0;

    if (w0lane) {
      if (q < NCHUNK - 1) {
        // Prefetch next chunk's patch into the other buffer.
        u32x4 g0; i32x8 g1; i32x4 g2, g3;
        uint64_t ga = feat_b + 4ull * (uint64_t)(q + 1) * CCHUNK * (FH * FW);
        tdm_build(&g0, &g1, &g2, &g3, ga, (q & 1) ? lds_b0 : lds_b1,
                  td0, td1, CCHUNK, PATCH, PATCH, CCHUNK, FW, FH * FW);
        tdm_load(g0, g1, g2, g3);
        // In-order retire: <=1 outstanding => T(q) done AND S(q-1) done.
        __builtin_amdgcn_s_wait_tensorcnt(1);
      } else {
        __builtin_amdgcn_s_wait_tensorcnt(0);
      }
    }
    __syncthreads();   // patch q visible to all waves; obuf free for rewrite

    // Thread t owns channel q*64+t: 49 bins x (2x2 samples x 2x2 taps) from LDS.
    const float* __restrict__ P =
        ((q & 1) ? &buf[1][0] : &buf[0][0]) + t * CHS;

    for (int ph = 0; ph < 7; ++ph) {
      const int sy0 = 2 * ph, sy1 = 2 * ph + 1;
      const int yA0 = (s_yl[sy0] - y0) * PATCH, yA1 = (s_yh[sy0] - y0) * PATCH;
      const int yB0 = (s_yl[sy1] - y0) * PATCH, yB1 = (s_yh[sy1] - y0) * PATCH;
      const float a0 = s_wy0[sy0], a1 = s_wy1[sy0];
      const float b0 = s_wy0[sy1], b1 = s_wy1[sy1];
      for (int pw = 0; pw < 7; ++pw) {
        const int sx0 = 2 * pw, sx1 = 2 * pw + 1;
        const int xA0 = s_xl[sx0] - x0, xA1 = s_xh[sx0] - x0;
        const int xB0 = s_xl[sx1] - x0, xB1 = s_xh[sx1] - x0;
        const float c0 = s_wx0[sx0], c1 = s_wx1[sx0];
        const float d0 = s_wx0[sx1], d1 = s_wx1[sx1];
        float acc;
        acc  = a0 * (c0 * P[yA0 + xA0] + c1 * P[yA0 + xA1])    // (sy0,sx0)
             + a1 * (c0 * P[yA1 + xA0] + c1 * P[yA1 + xA1]);
        acc += a0 * (d0 * P[yA0 + xB0] + d1 * P[yA0 + xB1])    // (sy0,sx1)
             + a1 * (d0 * P[yA1 + xB0] + d1 * P[yA1 + xB1]);
        acc += b0 * (c0 * P[yB0 + xA0] + c1 * P[yB0 + xA1])    // (sy1,sx0)
             + b1 * (c0 * P[yB1 + xA0] + c1 * P[yB1 + xA1]);
        acc += b0 * (d0 * P[yB0 + xB0] + d1 * P[yB0 + xB1])    // (sy1,sx1)
             + b1 * (d0 * P[yB1 + xB0] + d1 * P[yB1 + xB1]);
        obuf[t * NBIN + ph * 7 + pw] = acc * 0.25f;            // stride 49: no conflicts
      }
    }
    __syncthreads();   // obuf chunk complete across all waves

    // Stream the contiguous [64ch x 49] chunk out via TDM (1-D tile).
    if (w0lane) {
      u32x4 g0; i32x8 g1; i32x4 g2, g3;
      uint64_t ga = (uint64_t)(uintptr_t)out
                  + 4ull * ((uint64_t)k * C_TOTAL + (uint64_t)q * CCHUNK) * NBIN;
      const u32 n = CCHUNK * NBIN;   // 3136 elements
      tdm_build(&g0, &g1, &g2, &g3, ga, lds_ob, n, 1u, 1u,
                n, 0u, 0u, n, 0u);
      tdm_store(g0, g1, g2, g3);
      // Retirement of S(q) is guaranteed by next iteration's wait (<=1) or
      // by S_ENDPGM's implicit wait-idle on the last chunk.
    }
  }
}

extern "C" void kernel_launch(void* const* d_in, const int* in_sizes, int n_in,
                              void* d_out, int out_size, void* d_ws, size_t ws_size,
                              hipStream_t stream) {
  (void)n_in; (void)out_size; (void)d_ws; (void)ws_size;
  const float* feat = (const float*)d_in[0];
  const float* rois = (const float*)d_in[1];
  float* out = (float*)d_out;
  const int K = in_sizes[1] / 5;
  roi_align_tdm_kernel<<<dim3((unsigned)K), dim3(CCHUNK), 0, stream>>>(feat, rois, out, K);
}